// KDNet_82016695484689
// MI455X (gfx1250) — compile-verified
//
#include <hip/hip_runtime.h>

typedef float v2f __attribute__((ext_vector_type(2)));
typedef float v8f __attribute__((ext_vector_type(8)));

// Fused kd-conv layer:  Out = kd_gather_maxpool( relu(W @ X + bias) )   (mode 0)
// or plain GEMM+bias store for the FC head                              (mode 1)
//
// GEMM view: C[M,N] = W[M,K] * X[K,N],  N = B*D (always a multiple of 64),
// column n=(b,d), X[k,n] at X + b*K*D + k*D + d (reference layout [B,C_in,D]).
// D is a power of two -> Dlog2 shift/mask; all indices 32-bit.
//
// Grid: blockIdx.y = M-tile (16 rows), blockIdx.x = group of 8 N-tiles of 64
// columns (one per wave; 4 sub-tiles of 16 share the A fragment).
// The block's 16xK W tile is staged once into LDS, zero-padded in both M and
// K (to Kp4 = ceil4(K)), so the inner loop has no bounds selects at all.
__global__ __launch_bounds__(256) void kd_conv_wmma(
    const float* __restrict__ X, const float* __restrict__ Wt,
    const float* __restrict__ bias, const int* __restrict__ sel,
    float* __restrict__ Out, int M, int K, int Dlog2, int Bsz, int mode)
{
    extern __shared__ float sW[];             // 16 rows x Kp, Kp = Kp4+1 (odd)
    const int D   = 1 << Dlog2;
    const int Kp4 = (K + 3) & ~3;
    const int Kp  = Kp4 + 1;                  // odd stride: conflict-free A reads
    const int m0  = blockIdx.y << 4;

    // Cooperative stage of the W tile (before any exit path -> barrier-safe).
    for (int r = 0; r < 16; ++r) {
        const int m = m0 + r;
        const float* wr = Wt + (size_t)m * K;
        for (int k = threadIdx.x; k < Kp4; k += 256) {
            float w = 0.0f;
            if (m < M && k < K) w = wr[k];
            sW[r * Kp + k] = w;
        }
    }
    __syncthreads();

    const int lane = threadIdx.x & 31;
    const int ng   = (blockIdx.x << 3) + (threadIdx.x >> 5);
    const int n0   = ng << 6;

    // ISA 16x16x4 f32 fragment layout:
    //  A: lanes 0-15 hold (M=lane, K=0/1), lanes 16-31 hold (M=lane-16, K=2/3)
    //  B: mirrored (col = lane&15, rows K=0/1 low half, K=2/3 high half)
    //  C/D: vgpr r -> M = m0 + r + (lane<16 ? 0 : 8), N = n0 + (lane&15)
    const int half = lane >> 4;
    const int l16  = lane & 15;
    const int kOff = half << 1;
    const float* aRow = sW + l16 * Kp;        // LDS, zero-padded

    const float* xcol[4];
#pragma unroll
    for (int s = 0; s < 4; ++s) {
        const int n  = n0 + (s << 4) + l16;   // always < N (N % 64 == 0)
        const int bb = n >> Dlog2;
        const int dd = n & (D - 1);
        xcol[s] = X + (size_t)bb * ((size_t)K << Dlog2) + dd;
    }

    v8f acc[4] = { v8f{}, v8f{}, v8f{}, v8f{} };

    // Main K loop: full 4-wide tiles, zero bounds checks.
    const int kFull = K & ~3;
    for (int kt = 0; kt < kFull; kt += 4) {
        const int k0 = kt + kOff;
        v2f a;
        a.x = aRow[k0];
        a.y = aRow[k0 + 1];
#pragma unroll
        for (int s = 0; s < 4; ++s) {
            v2f b;
            b.x = xcol[s][(size_t)k0 << Dlog2];
            b.y = xcol[s][(size_t)(k0 + 1) << Dlog2];
            acc[s] = __builtin_amdgcn_wmma_f32_16x16x4_f32(
                false, a, false, b, (short)0, acc[s], false, false);
        }
        if (kt + 4 < kFull) {
#pragma unroll
            for (int s = 0; s < 4; ++s)
                __builtin_prefetch(xcol[s] + ((size_t)(k0 + 4) << Dlog2), 0, 1);
        }
    }
    // Remainder tile (only K=3 in layer 1). A side is zero-padded in LDS, so
    // the B index just gets clamped (its value is multiplied by 0).
    if (kFull < K) {
        const int k0 = kFull + kOff;
        const int c0 = min(k0, K - 1);
        const int c1 = min(k0 + 1, K - 1);
        v2f a;
        a.x = aRow[k0];
        a.y = aRow[k0 + 1];
#pragma unroll
        for (int s = 0; s < 4; ++s) {
            v2f b;
            b.x = xcol[s][(size_t)c0 << Dlog2];
            b.y = xcol[s][(size_t)c1 << Dlog2];
            acc[s] = __builtin_amdgcn_wmma_f32_16x16x4_f32(
                false, a, false, b, (short)0, acc[s], false, false);
        }
    }

    // Epilogue. mode 0: bias+relu, inverse-gather + pairwise max-pool via f32
    // atomic max (each pooled cell gets exactly 2 candidates; relu >= 0 so the
    // zero-initialized output is the identity). mode 1: bias + plain store.
    const int Dh = D >> 1;
    const int F = M / 3;
#pragma unroll
    for (int s = 0; s < 4; ++s) {
        const int n  = n0 + (s << 4) + l16;
        const int bb = n >> Dlog2;
        const int dp = n & (D - 1);
#pragma unroll
        for (int r = 0; r < 8; ++r) {
            const int o = m0 + r + (half << 3);
            if (o >= M) continue;
            float v = acc[s][r] + bias[o];
            if (mode == 0) {
                v = fmaxf(v, 0.0f);
                const int f = o / 3;               // magic-mul, no div
                const int q = o - 3 * f;
                const int t = (q << Dlog2) + dp;   // t in [0, 3D)
                const int dsrc = t / 3;            // unique source position
                const int rr = t - 3 * dsrc;
                if (sel[dsrc] == rr) {
                    float* dst = Out + (size_t)bb * (F * Dh)
                                     + (size_t)f * Dh + (dsrc >> 1);
                    atomicMax(dst, v);             // global_atomic_max_num_f32
                }
            } else {
                Out[(size_t)n * M + o] = v;
            }
        }
    }
}

__global__ __launch_bounds__(256) void kd_logsoftmax16(
    const float* __restrict__ logits, float* __restrict__ out, int rows)
{
    int r = blockIdx.x * blockDim.x + threadIdx.x;
    if (r >= rows) return;
    float v[16];
    float mx = -3.402823466e38f;
#pragma unroll
    for (int i = 0; i < 16; ++i) { v[i] = logits[r * 16 + i]; mx = fmaxf(mx, v[i]); }
    float s = 0.0f;
#pragma unroll
    for (int i = 0; i < 16; ++i) s += expf(v[i] - mx);
    const float l = mx + logf(s);
#pragma unroll
    for (int i = 0; i < 16; ++i) out[r * 16 + i] = v[i] - l;
}

extern "C" void kernel_launch(void* const* d_in, const int* in_sizes, int n_in,
                              void* d_out, int out_size, void* d_ws, size_t ws_size,
                              hipStream_t stream) {
    static const int DIMS[11] = {2048, 1024, 512, 256, 128, 64, 32, 16, 8, 4, 2};
    static const int INCH[11] = {3, 8, 32, 64, 64, 64, 128, 256, 512, 512, 512};
    static const int FEAT[11] = {8, 32, 64, 64, 64, 128, 256, 512, 512, 512, 1024};
    const int Bsz = 512;

    // d_in layout: 0:x, 1..11:c0..c10, 12+2i:W(i+1), 13+2i:b(i+1), 34:Wfc, 35:bfc
    // ws layout: two ping-pong X buffers (max pooled size 512*32*512 f32 = 32MB)
    //            + 32KB logits scratch  (total ~67MB)
    const size_t XBYTES = (size_t)512 * 32 * 512 * sizeof(float);
    float* X0     = (float*)d_ws;
    float* X1     = (float*)((char*)d_ws + XBYTES);
    float* logits = (float*)((char*)d_ws + 2 * XBYTES);

    const float* cur = (const float*)d_in[0];
    for (int i = 0; i < 11; ++i) {
        const int Dd = DIMS[i], Kc = INCH[i], F = FEAT[i], M = 3 * F;
        const int Dlog2 = __builtin_ctz((unsigned)Dd);
        float* nxt = (i & 1) ? X1 : X0;
        const size_t outBytes = (size_t)Bsz * F * (Dd / 2) * sizeof(float);
        hipMemsetAsync(nxt, 0, outBytes, stream);  // identity for relu'd max-pool

        const int N = Bsz * Dd;
        const int tilesM = (M + 15) / 16;
        const int nGroups = N / 64;                // exact: N % 64 == 0
        const int blocksX = nGroups / 8;           // exact: nGroups % 8 == 0
        const int Kp4 = (Kc + 3) & ~3;
        const size_t ldsBytes = (size_t)16 * (Kp4 + 1) * sizeof(float);
        dim3 grid(blocksX, tilesM);
        kd_conv_wmma<<<grid, 256, ldsBytes, stream>>>(
            cur, (const float*)d_in[12 + 2 * i], (const float*)d_in[13 + 2 * i],
            (const int*)d_in[1 + i], nxt, M, Kc, Dlog2, Bsz, /*mode=*/0);
        cur = nxt;
    }

    // FC head: M=16, K=1024, D=1 (Dlog2=0), N=512 -> 8 waves = 1 block
    {
        const size_t ldsBytes = (size_t)16 * (1024 + 1) * sizeof(float);
        dim3 grid(1, 1);
        kd_conv_wmma<<<grid, 256, ldsBytes, stream>>>(
            cur, (const float*)d_in[34], (const float*)d_in[35], nullptr,
            logits, /*M=*/16, /*K=*/1024, /*Dlog2=*/0, Bsz, /*mode=*/1);
    }

    kd_logsoftmax16<<<(512 + 255) / 256, 256, 0, stream>>>(logits, (float*)d_out, 512);

    (void)in_sizes; (void)n_in; (void)out_size; (void)ws_size;
}